// SGC_86311662780547
// MI455X (gfx1250) — compile-verified
//
#include <hip/hip_runtime.h>
#include <math.h>

typedef __attribute__((ext_vector_type(2))) float v2f;
typedef __attribute__((ext_vector_type(8))) float v8f;

#define DIN  128
#define DOUT 64
#define XP   132   // padded LDS pitch for 128-wide tiles (breaks bank conflicts)
#define W2P  68    // padded LDS pitch for 64-wide W2

// ---------------------------------------------------------------------------
// Hop kernel: one wave per edge. 32 lanes read the 128-float source row as
// float4 (coalesced 512B), scale by edge weight, atomic-add into dst row.
// Working set sits in the 192MB L2, so gather+atomic runs at L2 speed.
// ---------------------------------------------------------------------------
__global__ __launch_bounds__(256) void sgc_hop_kernel(
    const float* __restrict__ xin,
    const int*   __restrict__ src,
    const int*   __restrict__ dst,
    const float* __restrict__ ew,
    float*       __restrict__ xout,
    int n_edges)
{
    int tid  = blockIdx.x * 256 + threadIdx.x;
    int e    = tid >> 5;
    if (e >= n_edges) return;
    int lane = threadIdx.x & 31;
    int s    = src[e];
    int d    = dst[e];
    float w  = ew[e];
    float4 v = ((const float4*)(xin + (size_t)s * DIN))[lane];
    float* o = xout + (size_t)d * DIN + lane * 4;
    atomicAdd(o + 0, v.x * w);
    atomicAdd(o + 1, v.y * w);
    atomicAdd(o + 2, v.z * w);
    atomicAdd(o + 3, v.w * w);
}

// ---------------------------------------------------------------------------
// Fused dense transform: relu(x@W1+b1) @ W2 + b2 -> log_softmax.
// 8 waves / block, 16 rows per wave, fp32 WMMA 16x16x4 throughout.
// ---------------------------------------------------------------------------
__global__ __launch_bounds__(256) void sgc_transform_kernel(
    const float* __restrict__ x,    // [N,128] post-propagation
    const float* __restrict__ W1,   // [128,128]
    const float* __restrict__ b1,   // [128]
    const float* __restrict__ W2,   // [128,64]
    const float* __restrict__ b2,   // [64]
    float*       __restrict__ out,  // [N,64]
    int n)
{
    extern __shared__ float smem[];
    float* sW1 = smem;                 // [128][XP]
    float* sW2 = sW1 + 128 * XP;       // [128][W2P]
    float* sX  = sW2 + 128 * W2P;      // [128][XP], 16-row slab per wave

    const int tid     = threadIdx.x;
    const int wave    = tid >> 5;
    const int lane    = tid & 31;
    const int half    = lane >> 4;     // 0: lanes 0-15, 1: lanes 16-31
    const int l15     = lane & 15;
    const int rowBase = blockIdx.x * 128;

    // Cooperative stage of W1 [128x128] -> LDS (pitch XP)
    for (int i = tid; i < 128 * 32; i += 256) {
        int r = i >> 5, c4 = i & 31;
        float4 v = ((const float4*)W1)[i];
        *(float4*)(sW1 + r * XP + c4 * 4) = v;
    }
    // W2 [128x64] -> LDS (pitch W2P)
    for (int i = tid; i < 128 * 16; i += 256) {
        int r = i >> 4, c4 = i & 15;
        float4 v = ((const float4*)W2)[i];
        *(float4*)(sW2 + r * W2P + c4 * 4) = v;
    }
    // x tile [128x128] -> LDS, zero-padding out-of-range rows (tail block)
    for (int i = tid; i < 128 * 32; i += 256) {
        int r = i >> 5, c4 = i & 31;
        int gr = rowBase + r;
        float4 v = {0.f, 0.f, 0.f, 0.f};
        if (gr < n) v = ((const float4*)x)[(size_t)gr * 32 + c4];
        *(float4*)(sX + r * XP + c4 * 4) = v;
    }
    __syncthreads();

    float* myX = sX + wave * 16 * XP;  // this wave's private 16-row slab

    // ---- lin1: h = relu(x @ W1 + b1); 8 column tiles of 16 ----
    v8f hacc[8];
    #pragma unroll
    for (int ct = 0; ct < 8; ++ct) {
        v8f acc = {};
        #pragma unroll
        for (int k = 0; k < DIN; k += 4) {
            v2f a, b;
            // A 16x4 fragment: lane holds row M=l15, K pair (2*half, 2*half+1)
            a.x = myX[l15 * XP + k + 2 * half];
            a.y = myX[l15 * XP + k + 2 * half + 1];
            // B 4x16 fragment: lane holds column N=ct*16+l15, same K pair
            b.x = sW1[(k + 2 * half)     * XP + ct * 16 + l15];
            b.y = sW1[(k + 2 * half + 1) * XP + ct * 16 + l15];
            acc = __builtin_amdgcn_wmma_f32_16x16x4_f32(
                false, a, false, b, (short)0, acc, false, false);
        }
        float bias = b1[ct * 16 + l15];
        #pragma unroll
        for (int r = 0; r < 8; ++r) {
            float hv = acc[r] + bias;       // C/D layout: row r+8*half, col ct*16+l15
            acc[r] = hv > 0.f ? hv : 0.f;   // relu
        }
        hacc[ct] = acc;
    }
    // Overwrite this wave's x slab with h (all x reads complete; region private)
    #pragma unroll
    for (int ct = 0; ct < 8; ++ct)
        #pragma unroll
        for (int r = 0; r < 8; ++r)
            myX[(r + 8 * half) * XP + ct * 16 + l15] = hacc[ct][r];

    // ---- lin2: logits = h @ W2 + b2; 4 column tiles ----
    v8f lacc[4];
    #pragma unroll
    for (int ct = 0; ct < 4; ++ct) {
        v8f acc = {};
        #pragma unroll
        for (int k = 0; k < DIN; k += 4) {
            v2f a, b;
            a.x = myX[l15 * XP + k + 2 * half];
            a.y = myX[l15 * XP + k + 2 * half + 1];
            b.x = sW2[(k + 2 * half)     * W2P + ct * 16 + l15];
            b.y = sW2[(k + 2 * half + 1) * W2P + ct * 16 + l15];
            acc = __builtin_amdgcn_wmma_f32_16x16x4_f32(
                false, a, false, b, (short)0, acc, false, false);
        }
        float bias = b2[ct * 16 + l15];
        #pragma unroll
        for (int r = 0; r < 8; ++r) acc[r] += bias;
        lacc[ct] = acc;
    }
    // Stage logits [16][64] into the slab (h fully consumed by now)
    #pragma unroll
    for (int ct = 0; ct < 4; ++ct)
        #pragma unroll
        for (int r = 0; r < 8; ++r)
            myX[(r + 8 * half) * 64 + ct * 16 + l15] = lacc[ct][r];

    // ---- row-wise log_softmax over 64 logits (wave32 shfl reduction) ----
    for (int r = 0; r < 16; ++r) {
        float v0 = myX[r * 64 + lane];
        float v1 = myX[r * 64 + 32 + lane];
        float m = fmaxf(v0, v1);
        #pragma unroll
        for (int off = 16; off > 0; off >>= 1)
            m = fmaxf(m, __shfl_xor(m, off, 32));
        float s = __expf(v0 - m) + __expf(v1 - m);
        #pragma unroll
        for (int off = 16; off > 0; off >>= 1)
            s += __shfl_xor(s, off, 32);
        float lse = m + __logf(s);
        int grow = rowBase + wave * 16 + r;
        if (grow < n) {
            out[(size_t)grow * 64 + lane]      = v0 - lse;
            out[(size_t)grow * 64 + 32 + lane] = v1 - lse;
        }
    }
}

// ---------------------------------------------------------------------------
extern "C" void kernel_launch(void* const* d_in, const int* in_sizes, int n_in,
                              void* d_out, int out_size, void* d_ws, size_t ws_size,
                              hipStream_t stream) {
    (void)n_in; (void)out_size; (void)ws_size;
    const float* x  = (const float*)d_in[0];
    const int*   ei = (const int*)  d_in[1];   // [2,E]
    const float* ew = (const float*)d_in[2];
    // d_in[3] = is_cluster scalar (0 -> non-cluster path; ignored)
    const float* W1 = (const float*)d_in[4];
    const float* b1 = (const float*)d_in[5];
    const float* W2 = (const float*)d_in[6];
    const float* b2 = (const float*)d_in[7];
    float* out = (float*)d_out;

    int N = in_sizes[0] / DIN;
    int E = in_sizes[1] / 2;
    const int* src = ei;
    const int* dst = ei + E;

    float* buf1 = (float*)d_ws;                 // N*128 fp32
    float* buf2 = buf1 + (size_t)N * DIN;       // N*128 fp32
    hipMemsetAsync(d_ws, 0, (size_t)2 * N * DIN * sizeof(float), stream);

    int hopBlocks = (E * 32 + 255) / 256;       // one wave per edge
    sgc_hop_kernel<<<hopBlocks, 256, 0, stream>>>(x,    src, dst, ew, buf1, E);
    sgc_hop_kernel<<<hopBlocks, 256, 0, stream>>>(buf1, src, dst, ew, buf2, E);

    size_t shmem = (size_t)(128 * XP * 2 + 128 * W2P) * sizeof(float); // ~166KB of the 320KB LDS
    int tBlocks = (N + 127) / 128;
    sgc_transform_kernel<<<tBlocks, 256, shmem, stream>>>(buf2, W1, b1, W2, b2, out, N);
}